// FRNOD_34772055229178
// MI455X (gfx1250) — compile-verified
//
#include <hip/hip_runtime.h>
#include <hip/hip_bf16.h>

typedef float v2f __attribute__((ext_vector_type(2)));
typedef float v8f __attribute__((ext_vector_type(8)));

#define WAY   2
#define RES   49
#define CH    640
#define NQ    49049           // 1001 * 49
#define NBOX  1001
#define EPAD  49056           // NQ padded to multiple of 32
#define KSTEPS 160            // 640 / 4
#define NTILES 7              // 112 columns = 7 * 16 (98 used)

// workspace layout (float units)
#define SCAL_OFF   0          // [0]=a=(rho^2-2rho), [1]=b=-rho^2*lam, [2]=lam, [3]=rho
#define GINV_OFF   16         // 2 ways * 2401, stride 2432
#define GINV_STRIDE 2432
#define BPACK_OFF  8192       // 160*7*32 lanes * 2 floats = 71680
#define E_OFF      (8192 + 71680)   // 2 * EPAD = 98112 floats

// ---------------------------------------------------------------------------
// Kernel 1: scalars + G = S S^T + lam I (49x49, per way) + Gauss-Jordan inverse
// ---------------------------------------------------------------------------
__global__ void __launch_bounds__(128) frn_prep(const float* __restrict__ support,
                                                const float* __restrict__ r,
                                                float* __restrict__ ws) {
  __shared__ float Aug[RES * 98];
  __shared__ float fcol[RES];
  __shared__ float s_lam;
  const int t = threadIdx.x;
  const int NT = blockDim.x;

  if (t == 0) {
    float lam = ((float)RES / (float)CH) * __expf(r[0]) + 1e-6f;
    float rho = __expf(r[1]);
    ws[SCAL_OFF + 0] = rho * rho - 2.0f * rho;
    ws[SCAL_OFF + 1] = -rho * rho * lam;
    ws[SCAL_OFF + 2] = lam;
    ws[SCAL_OFF + 3] = rho;
    s_lam = lam;
  }
  __syncthreads();
  const float lam = s_lam;

  for (int w = 0; w < WAY; ++w) {
    const float* S = support + (size_t)w * RES * CH;
    // init augmented [G | I]
    for (int idx = t; idx < RES * 98; idx += NT) {
      int i = idx / 98, c = idx % 98;
      float v;
      if (c < RES) {
        const float* a = S + (size_t)i * CH;
        const float* b = S + (size_t)c * CH;
        float acc = 0.0f;
        for (int k = 0; k < CH; ++k) acc = fmaf(a[k], b[k], acc);
        if (c == i) acc += lam;
        v = acc;
      } else {
        v = ((c - RES) == i) ? 1.0f : 0.0f;
      }
      Aug[idx] = v;
    }
    __syncthreads();

    // Gauss-Jordan (SPD: no pivoting needed)
    for (int p = 0; p < RES; ++p) {
      for (int i = t; i < RES; i += NT) fcol[i] = Aug[i * 98 + p];
      __syncthreads();
      float pivinv = 1.0f / fcol[p];
      for (int c = t; c < 98; c += NT) Aug[p * 98 + c] *= pivinv;
      __syncthreads();
      for (int idx = t; idx < RES * 98; idx += NT) {
        int i = idx / 98, c = idx % 98;
        if (i != p) Aug[idx] = fmaf(-fcol[i], Aug[p * 98 + c], Aug[idx]);
      }
      __syncthreads();
    }
    // write G^{-1}
    for (int idx = t; idx < RES * RES; idx += NT) {
      int i = idx / RES, c = idx % RES;
      ws[GINV_OFF + w * GINV_STRIDE + idx] = Aug[i * 98 + 49 + c];
    }
    __syncthreads();
  }
}

// ---------------------------------------------------------------------------
// Kernel 2: pre-pack B = S^T (640 x 112, cols 0..48 way0, 49..97 way1, rest 0)
// into the per-lane V_WMMA_F32_16X16X4_F32 B-operand layout:
//   VGPR0 = B[k0 + 2*(lane/16) + 0][ntile*16 + lane%16]
//   VGPR1 = B[k0 + 2*(lane/16) + 1][ntile*16 + lane%16]
// stored as float2 at bpack[(kstep*7 + ntile)*32 + lane].
// ---------------------------------------------------------------------------
__global__ void frn_pack(const float* __restrict__ support, float* __restrict__ ws) {
  int idx = blockIdx.x * blockDim.x + threadIdx.x;
  if (idx >= KSTEPS * NTILES * 32) return;
  int lane = idx & 31;
  int nt = (idx >> 5) % NTILES;
  int ks = idx / (NTILES * 32);
  int col = nt * 16 + (lane & 15);
  int k0 = ks * 4 + 2 * (lane >> 4);
  float v0 = 0.0f, v1 = 0.0f;
  if (col < WAY * RES) {
    int w = (col < RES) ? 0 : 1;
    int rr = col - w * RES;
    const float* Sp = support + ((size_t)w * RES + rr) * CH;
    v0 = Sp[k0];
    v1 = Sp[k0 + 1];
  }
  ws[BPACK_OFF + (size_t)idx * 2 + 0] = v0;
  ws[BPACK_OFF + (size_t)idx * 2 + 1] = v1;
}

// ---------------------------------------------------------------------------
// Kernel 3: main fused pass.  Per wave: 16 query rows.
//   T[16x112] = Q_tile[16x640] @ B[640x112]  via v_wmma_f32_16x16x4_f32
//   then per (row m, way w) lane: z = Ginv_w t,  e = a*(t.z) + b*(z.z) + |q|^2
// ---------------------------------------------------------------------------
__global__ void __launch_bounds__(128) frn_main(const float* __restrict__ query,
                                                const float* __restrict__ ws,
                                                float* __restrict__ E) {
  __shared__ float Ginv_s[WAY * RES * RES];      // 4802 floats
  __shared__ float T_s[4][16][114];              // per-wave T tile (padded stride)
  __shared__ float qn_s[4][32];

  const float* bpack = ws + BPACK_OFF;
  const float* ginv = ws + GINV_OFF;
  const int tid = threadIdx.x;
  const int lane = tid & 31;
  const int wave = tid >> 5;

  // cooperative Ginv load into LDS (both ways)
  for (int i = tid; i < RES * RES; i += 128) {
    Ginv_s[i] = ginv[i];
    Ginv_s[RES * RES + i] = ginv[GINV_STRIDE + i];
  }
  const float ca = ws[SCAL_OFF + 0];
  const float cb = ws[SCAL_OFF + 1];
  __syncthreads();

  const int row0 = (blockIdx.x * 4 + wave) * 16;
  const int m = lane & 15;
  const int kh = lane >> 4;
  long rowm = (long)row0 + m;
  if (rowm > NQ - 1) rowm = NQ - 1;              // clamp: uniform flow, guard store later
  const float* qp = query + rowm * (long)CH + kh * 2;
  const v2f* bp = (const v2f*)bpack;

  v8f acc[NTILES];
#pragma unroll
  for (int nt = 0; nt < NTILES; ++nt) acc[nt] = (v8f){};
  float qsq = 0.0f;

#pragma unroll 2
  for (int k = 0; k < KSTEPS; ++k) {
    v2f a = *(const v2f*)(qp + k * 4);
    qsq = fmaf(a[0], a[0], fmaf(a[1], a[1], qsq));
    const v2f* bk = bp + (size_t)k * (NTILES * 32) + lane;
#pragma unroll
    for (int nt = 0; nt < NTILES; ++nt) {
      v2f b = bk[nt * 32];
      acc[nt] = __builtin_amdgcn_wmma_f32_16x16x4_f32(
          /*neg_a=*/false, a, /*neg_b=*/false, b,
          /*c_mod=*/(short)0, acc[nt], /*reuse_a=*/false, /*reuse_b=*/false);
    }
  }

  // spill C tiles to LDS: C layout lane L, VGPR v -> row = v + 8*(L/16), col = nt*16 + L%16
#pragma unroll
  for (int nt = 0; nt < NTILES; ++nt)
#pragma unroll
    for (int v = 0; v < 8; ++v)
      T_s[wave][v + 8 * kh][nt * 16 + m] = acc[nt][v];
  qn_s[wave][lane] = qsq;
  __syncthreads();

  // phase 2: lane = m + 16*w handles (query m, way w)
  const int w = lane >> 4;
  const float qnorm = qn_s[wave][m] + qn_s[wave][m + 16];
  const float* tv = &T_s[wave][m][w * RES];
  const float* gw = &Ginv_s[w * RES * RES];
  float tz = 0.0f, zz = 0.0f;
  for (int i = 0; i < RES; ++i) {
    const float* gr = gw + i * RES;
    float zi = 0.0f;
#pragma unroll 7
    for (int j = 0; j < RES; ++j) zi = fmaf(gr[j], tv[j], zi);
    tz = fmaf(zi, tv[i], tz);
    zz = fmaf(zi, zi, zz);
  }
  float e = fmaf(ca, tz, fmaf(cb, zz, qnorm));
  long row = (long)row0 + m;
  if (row < NQ) E[(size_t)w * EPAD + row] = e;
}

// ---------------------------------------------------------------------------
// Kernel 4: per-box mean over 49, log-softmax over 2 ways, mean over boxes.
// Single workgroup tree reduction -> deterministic.
// ---------------------------------------------------------------------------
__global__ void __launch_bounds__(256) frn_reduce(const float* __restrict__ ws,
                                                  const float* __restrict__ scale,
                                                  float* __restrict__ out) {
  __shared__ float red[256];
  const float* E = ws + E_OFF;
  const float sc = scale[0];
  float lsum = 0.0f;
  for (int b = threadIdx.x; b < NBOX; b += 256) {
    float s0 = 0.0f, s1 = 0.0f;
    int base = b * RES;
    for (int rr = 0; rr < RES; ++rr) {
      s0 += E[base + rr];
      s1 += E[EPAD + base + rr];
    }
    float l0 = (-s0 / (float)RES) * sc;
    float l1 = (-s1 / (float)RES) * sc;
    float mx = fmaxf(l0, l1);
    float lse = mx + __logf(__expf(l0 - mx) + __expf(l1 - mx));
    lsum += (lse - l0);   // = -logp[:,0]
  }
  red[threadIdx.x] = lsum;
  __syncthreads();
  for (int s = 128; s > 0; s >>= 1) {
    if (threadIdx.x < s) red[threadIdx.x] += red[threadIdx.x + s];
    __syncthreads();
  }
  if (threadIdx.x == 0) out[0] = red[0] / (float)NBOX;
}

// ---------------------------------------------------------------------------
extern "C" void kernel_launch(void* const* d_in, const int* in_sizes, int n_in,
                              void* d_out, int out_size, void* d_ws, size_t ws_size,
                              hipStream_t stream) {
  const float* support = (const float*)d_in[0];   // [2,49,640]
  const float* query   = (const float*)d_in[1];   // [49049,640]
  const float* r       = (const float*)d_in[2];   // [2]
  const float* scale   = (const float*)d_in[3];   // [1]
  float* ws = (float*)d_ws;
  float* out = (float*)d_out;

  frn_prep<<<1, 128, 0, stream>>>(support, r, ws);
  frn_pack<<<(KSTEPS * NTILES * 32 + 255) / 256, 256, 0, stream>>>(support, ws);
  frn_main<<<(NQ + 63) / 64, 128, 0, stream>>>(query, ws, ws + E_OFF);
  frn_reduce<<<1, 256, 0, stream>>>(ws, scale, out);
}